// GeATLayer_43061342110000
// MI455X (gfx1250) — compile-verified
//
#include <hip/hip_runtime.h>

typedef float v2f __attribute__((ext_vector_type(2)));
typedef float v8f __attribute__((ext_vector_type(8)));
typedef int v4i_vs __attribute__((vector_size(4 * sizeof(int))));

#define NB 256
#define MM 128
#define DD 64
#define HH 8
#define DHH 512
#define NEG_SLOPE 0.2f

// ---------------------------------------------------------------------------
// CDNA5 async global->LDS path (ASYNCcnt). Guarded so compilation never
// regresses if the toolchain lacks the builtin; fallback is a VGPR round-trip.
// Builtin signature (from hipcc diagnostic): (v4i AS1*, v4i AS3*, Iint, Iint)
// ---------------------------------------------------------------------------
#if defined(__has_builtin)
#if __has_builtin(__builtin_amdgcn_global_load_async_to_lds_b128)
#define HAVE_ASYNC_LDS 1
#endif
#endif
#ifndef HAVE_ASYNC_LDS
#define HAVE_ASYNC_LDS 0
#endif

__device__ __forceinline__ void async_cp16(void* l, const void* g) {
#if HAVE_ASYNC_LDS
  __builtin_amdgcn_global_load_async_to_lds_b128(
      (__attribute__((address_space(1))) v4i_vs*)(g),
      (__attribute__((address_space(3))) v4i_vs*)(l), 0, 0);
#else
  *(float4*)l = *(const float4*)g;
#endif
}

#if HAVE_ASYNC_LDS
#if __has_builtin(__builtin_amdgcn_s_wait_asynccnt)
#define ASYNC_WAIT(N) __builtin_amdgcn_s_wait_asynccnt(N)
#else
#define ASYNC_WAIT(N) asm volatile("s_wait_asynccnt %0" ::"i"(N) : "memory")
#endif
#else
#define ASYNC_WAIT(N) \
  do {                \
  } while (0)
#endif

__device__ __forceinline__ v8f wmma_f32(v2f a, v2f b, v8f c) {
  // D = A(16x4) * B(4x16) + C(16x16), fp32
  return __builtin_amdgcn_wmma_f32_16x16x4_f32(false, a, false, b, (short)0, c,
                                               false, false);
}

// ---------------------------------------------------------------------------
// Kernel 1: per (batch, head) fused attention.
//   Q = (X Wq_h + bq_h) * 1/sqrt(D); K = X Wk_h + bk_h; V = X Wv_h + bv_h
//   S = Q K^T + edge_bias[edges]; leaky; mask(edge==0 -> -1e9); softmax
//   O = S V  ->  ws[b, m, h, d]   (layout [B, M, H*D])
// ---------------------------------------------------------------------------
__global__ __launch_bounds__(256) void geat_attn_kernel(
    const float* __restrict__ Xg, const int* __restrict__ edges,
    const float* __restrict__ Wq, const float* __restrict__ bq,
    const float* __restrict__ Wk, const float* __restrict__ bk,
    const float* __restrict__ Wv, const float* __restrict__ bv,
    const float* __restrict__ ebias, float* __restrict__ ws) {
  __shared__ float smem[40960];  // 160 KB
  float* sX = smem;              // 8192 floats   [128 x 64]
  float* sW = smem + 8192;       // 4096 floats   [64 x 64]
  float* sS = smem;              // 16384 floats  [128 x 128] (aliases sX/sW)
  float* sQ = smem + 16384;      // 8192
  float* sK = smem + 24576;      // 8192
  float* sV = smem + 32768;      // 8192

  const int tid = threadIdx.x;
  const int wv = tid >> 5;  // wave id 0..7
  const int lane = tid & 31;
  const int hl = lane >> 4;  // half-wave select
  const int r = lane & 15;
  const int b = blockIdx.x >> 3;
  const int h = blockIdx.x & 7;

  // ---- async load X_b [128 x 64] ----
  {
    const float4* src = (const float4*)(Xg + (size_t)b * MM * DD);
    float4* dst = (float4*)sX;
    for (int i = tid; i < MM * DD / 4; i += 256) async_cp16(dst + i, src + i);
  }

  const float* Wt[3] = {Wq, Wk, Wv};
  const float* bt[3] = {bq, bk, bv};
  const int qkvOff[3] = {16384, 24576, 32768};  // integer offsets, no LDS-ptr
                                                // arrays (lld addrspacecast)

  // ---- QKV projections: [128x64] = [128x64] @ [64x64] ----
  for (int p = 0; p < 3; ++p) {
    {  // async load head slice of W: sW[d][j] = W[d][h*64 + j]
      const float4* wsrc = (const float4*)Wt[p];
      float4* wdst = (float4*)sW;
      for (int i = tid; i < DD * DD / 4; i += 256) {
        int d = i >> 4, jj = i & 15;
        async_cp16(wdst + i, wsrc + d * (DHH / 4) + h * (DD / 4) + jj);
      }
    }
    ASYNC_WAIT(0);
    __syncthreads();
    float* dst = smem + qkvOff[p];
    const float* bias = bt[p];
    const float mul = (p == 0) ? 0.125f : 1.0f;  // fold 1/sqrt(64) into Q
    for (int t = wv; t < 32; t += 8) {           // 8x4 tiles of 16x16
      const int m0 = (t >> 2) * 16, n0 = (t & 3) * 16;
      v8f acc = {0.f, 0.f, 0.f, 0.f, 0.f, 0.f, 0.f, 0.f};
#pragma unroll
      for (int k0 = 0; k0 < DD; k0 += 4) {
        v2f a, bb;
        const float* ap = sX + (m0 + r) * DD + k0 + 2 * hl;
        a.x = ap[0];
        a.y = ap[1];
        bb.x = sW[(k0 + 2 * hl) * DD + n0 + r];
        bb.y = sW[(k0 + 2 * hl + 1) * DD + n0 + r];
        acc = wmma_f32(a, bb, acc);
      }
      const int n = n0 + r;
      const float bsv = bias[h * DD + n];
#pragma unroll
      for (int j = 0; j < 8; ++j) {
        const int m = m0 + hl * 8 + j;
        dst[m * DD + n] = (acc[j] + bsv) * mul;
      }
    }
    __syncthreads();
  }

  // ---- scores S = Q @ K^T, fused edge bias + leaky relu + mask ----
  for (int t = wv; t < 64; t += 8) {  // 8x8 tiles
    const int m0 = (t >> 3) * 16, n0 = (t & 7) * 16;
    v8f acc = {0.f, 0.f, 0.f, 0.f, 0.f, 0.f, 0.f, 0.f};
#pragma unroll
    for (int k0 = 0; k0 < DD; k0 += 4) {
      v2f a, bb;
      const float* ap = sQ + (m0 + r) * DD + k0 + 2 * hl;
      a.x = ap[0];
      a.y = ap[1];
      const float* bp_ = sK + (n0 + r) * DD + k0 + 2 * hl;  // K^T
      bb.x = bp_[0];
      bb.y = bp_[1];
      acc = wmma_f32(a, bb, acc);
    }
    const int n = n0 + r;
#pragma unroll
    for (int j = 0; j < 8; ++j) {
      const int m = m0 + hl * 8 + j;
      const int e = edges[((size_t)b * MM + m) * MM + n];
      float s = acc[j] + ebias[e * HH + h];
      s = (s > 0.f) ? s : NEG_SLOPE * s;
      if (e == 0) s = -1e9f;
      sS[m * MM + n] = s;
    }
  }
  __syncthreads();

  // ---- row softmax ----
  if (tid < MM) {
    float* row = sS + tid * MM;
    float mx = -3.4e38f;
#pragma unroll 4
    for (int n = 0; n < MM; ++n) mx = fmaxf(mx, row[n]);
    float sum = 0.f;
#pragma unroll 4
    for (int n = 0; n < MM; ++n) {
      const float e = __expf(row[n] - mx);
      row[n] = e;
      sum += e;
    }
    const float inv = 1.f / sum;
#pragma unroll 4
    for (int n = 0; n < MM; ++n) row[n] *= inv;
  }
  __syncthreads();

  // ---- O = P @ V -> ws[b, m, h*64 + d] ----
  for (int t = wv; t < 32; t += 8) {
    const int m0 = (t >> 2) * 16, n0 = (t & 3) * 16;
    v8f acc = {0.f, 0.f, 0.f, 0.f, 0.f, 0.f, 0.f, 0.f};
#pragma unroll
    for (int k0 = 0; k0 < MM; k0 += 4) {
      v2f a, bb;
      const float* ap = sS + (m0 + r) * MM + k0 + 2 * hl;
      a.x = ap[0];
      a.y = ap[1];
      bb.x = sV[(k0 + 2 * hl) * DD + n0 + r];
      bb.y = sV[(k0 + 2 * hl + 1) * DD + n0 + r];
      acc = wmma_f32(a, bb, acc);
    }
    const int n = n0 + r;
#pragma unroll
    for (int j = 0; j < 8; ++j) {
      const int m = m0 + hl * 8 + j;
      ws[((size_t)(b * MM + m)) * DHH + h * DD + n] = acc[j];
    }
  }
}

// ---------------------------------------------------------------------------
// Kernel 2: out[32768 x 64] = ws[32768 x 512] @ Wp[512 x 64] + bp
// Double-buffered K-chunks via async LDS loads (24 async ops/wave/chunk).
// Ping-pong buffer selected by integer offset (no LDS-pointer arrays).
// ---------------------------------------------------------------------------
__device__ __forceinline__ void proj_load_chunk(const float4* asrc,
                                                const float4* wsrc, float* buf,
                                                int row0, int kk, int tid) {
  float4* adst = (float4*)buf;           // A chunk [128 x 128] at offset 0
  float4* wdst = (float4*)(buf + 16384);  // Wp chunk [128 x 64] after A
  for (int i = tid; i < 4096; i += 256) {
    const int m = i >> 5, c = i & 31;
    async_cp16(adst + i, asrc + (size_t)(row0 + m) * (DHH / 4) + (kk >> 2) + c);
  }
  for (int i = tid; i < 2048; i += 256) {
    const int d = i >> 4, jj = i & 15;
    async_cp16(wdst + i, wsrc + (size_t)(kk + d) * (DD / 4) + jj);
  }
}

__global__ __launch_bounds__(256) void geat_proj_kernel(
    const float* __restrict__ wsin, const float* __restrict__ Wp,
    const float* __restrict__ bp, float* __restrict__ out) {
  __shared__ float smem[49152];  // 192 KB, double buffered (24576 floats each)

  const int tid = threadIdx.x;
  const int wv = tid >> 5;
  const int lane = tid & 31;
  const int hl = lane >> 4;
  const int r = lane & 15;
  const int row0 = blockIdx.x * 128;
  const float4* asrc = (const float4*)wsin;
  const float4* wsrc = (const float4*)Wp;

  v8f acc[4];
#pragma unroll
  for (int i = 0; i < 4; ++i)
    acc[i] = (v8f){0.f, 0.f, 0.f, 0.f, 0.f, 0.f, 0.f, 0.f};

  proj_load_chunk(asrc, wsrc, smem, row0, 0, tid);

  for (int c = 0; c < 4; ++c) {
    if (c < 3) {
      // prefetch next chunk, then wait until only those 24 ops are in flight
      proj_load_chunk(asrc, wsrc, smem + ((c + 1) & 1) * 24576, row0,
                      (c + 1) * 128, tid);
      ASYNC_WAIT(24);
    } else {
      ASYNC_WAIT(0);
    }
    __syncthreads();
    const float* cA = smem + (c & 1) * 24576;
    const float* cW = cA + 16384;
#pragma unroll
    for (int i = 0; i < 4; ++i) {
      const int t = wv + i * 8;
      const int m0 = (t >> 2) * 16, n0 = (t & 3) * 16;
      v8f a8 = acc[i];
#pragma unroll
      for (int k0 = 0; k0 < 128; k0 += 4) {
        v2f a, bb;
        const float* ap = cA + (m0 + r) * 128 + k0 + 2 * hl;
        a.x = ap[0];
        a.y = ap[1];
        bb.x = cW[(k0 + 2 * hl) * DD + n0 + r];
        bb.y = cW[(k0 + 2 * hl + 1) * DD + n0 + r];
        a8 = wmma_f32(a, bb, a8);
      }
      acc[i] = a8;
    }
    __syncthreads();
  }

#pragma unroll
  for (int i = 0; i < 4; ++i) {
    const int t = wv + i * 8;
    const int m0 = (t >> 2) * 16, n0 = (t & 3) * 16;
    const int n = n0 + r;
    const float bsv = bp[n];
#pragma unroll
    for (int j = 0; j < 8; ++j) {
      out[(size_t)(row0 + m0 + hl * 8 + j) * DD + n] = acc[i][j] + bsv;
    }
  }
}

extern "C" void kernel_launch(void* const* d_in, const int* in_sizes, int n_in,
                              void* d_out, int out_size, void* d_ws,
                              size_t ws_size, hipStream_t stream) {
  (void)in_sizes;
  (void)n_in;
  (void)out_size;
  (void)ws_size;
  const float* X = (const float*)d_in[0];
  const int* edges = (const int*)d_in[1];
  const float* Wq = (const float*)d_in[2];
  const float* bq = (const float*)d_in[3];
  const float* Wk = (const float*)d_in[4];
  const float* bk = (const float*)d_in[5];
  const float* Wv = (const float*)d_in[6];
  const float* bv = (const float*)d_in[7];
  const float* eb = (const float*)d_in[8];
  const float* Wp = (const float*)d_in[9];
  const float* bp = (const float*)d_in[10];
  float* out = (float*)d_out;
  float* ws = (float*)d_ws;  // needs B*M*H*D*4 = 64 MB

  geat_attn_kernel<<<dim3(NB * HH), dim3(256), 0, stream>>>(
      X, edges, Wq, bq, Wk, bk, Wv, bv, eb, ws);
  geat_proj_kernel<<<dim3((NB * MM) / 128), dim3(256), 0, stream>>>(ws, Wp, bp,
                                                                    out);
}